// RoiPoolingLayer_49392123904363
// MI455X (gfx1250) — compile-verified
//
#include <hip/hip_runtime.h>
#include <hip/hip_bf16.h>
#include <stdint.h>

// ROI bilinear pooling (TF1 resize semantics), MI455X / gfx1250.
//
// Roofline: ~12.85 MB written + <=51 MB gathered = ~64 MB -> ~2.8 us floor
// at 23.3 TB/s; ~29 MFLOP -> 0.45 flop/byte => purely memory-bound. WMMA
// reformulation (W[49,196] @ V per ROI) inflates FLOPs ~65x without cutting
// bytes, so the CDNA5 feature that fits is the async global->LDS gather path.
//
// Strategy: 1 block per (roi, py); 7 waves = 7 px columns; 32 lanes = 32
// float4 channel chunks (128 ch). Each wave fires its 4 neighbor-row gathers
// (4 x 512 B, b128 per lane) as ASYNC loads into LDS (ASYNCcnt, no dest
// VGPRs), waits once on s_wait_asynccnt, blends from LDS, and stores a
// coalesced 512 B output row non-temporally (output is write-once).

#define POOL   7
#define IMG_H  1024
#define IMG_W  1024
#define NCH    128          // channels

typedef float v4f __attribute__((ext_vector_type(4)));  // native vector: OK for
                                                        // __builtin_nontemporal_store

__device__ __forceinline__ void axis_coords(int start, int size, int j, int limit,
                                            int& lo_abs, int& hi_abs, float& frac)
{
    // TF1 bilinear, align_corners=False: src = j * size / POOL
    float scale = (float)size * (1.0f / (float)POOL);
    float src   = (float)j * scale;
    int lo = (int)floorf(src);
    lo = min(max(lo, 0), size - 1);
    int hi = min(lo + 1, size - 1);
    frac = src - (float)lo;
    lo_abs = min(max(start + lo, 0), limit - 1);
    hi_abs = min(max(start + hi, 0), limit - 1);
}

__device__ __forceinline__ uint32_t lds_offset(const void* p)
{
    // Flat->LDS mapping truncates to addr[31:0]; low 32 bits of a generic
    // pointer to __shared__ are the LDS byte offset.
    return (uint32_t)(uintptr_t)p;
}

__device__ __forceinline__ void async_load_b128_to_lds(uint32_t lds_off, const void* gaddr)
{
    // RT near / NT far: intra-block rehits (shared x_lo/x_hi columns) stay in
    // the near cache, but scattered ROI gathers don't churn global L2.
    asm volatile("global_load_async_to_lds_b128 %0, %1, off th:TH_LOAD_RT_NT"
                 :
                 : "v"(lds_off), "v"((uint64_t)(uintptr_t)gaddr)
                 : "memory");
}

__global__ __launch_bounds__(POOL * 32)
void roi_bilinear_async_kernel(const float* __restrict__ img,
                               const float* __restrict__ rois,
                               float* __restrict__ out)
{
    // smem[px][neighbor][lane] : 7 * 4 * 32 * 16B = 14336 B
    __shared__ v4f smem[POOL][4][32];

    const int roi  = blockIdx.x;
    const int py   = blockIdx.y;
    const int px   = threadIdx.x >> 5;   // wave id = output x position
    const int lane = threadIdx.x & 31;   // float4 channel chunk

    const float* r = rois + (size_t)roi * 4;
    const int x1 = (int)r[0];
    const int y1 = (int)r[1];
    const int rw = (int)r[2];
    const int rh = (int)r[3];

    int ylo, yhi; float fy;
    axis_coords(y1, rh, py, IMG_H, ylo, yhi, fy);
    int xlo, xhi; float fx;
    axis_coords(x1, rw, px, IMG_W, xlo, xhi, fx);

    const int lc = lane * 4; // channel offset (floats)
    const float* p00 = img + ((size_t)(ylo * IMG_W + xlo) * NCH + lc);
    const float* p01 = img + ((size_t)(ylo * IMG_W + xhi) * NCH + lc);
    const float* p10 = img + ((size_t)(yhi * IMG_W + xlo) * NCH + lc);
    const float* p11 = img + ((size_t)(yhi * IMG_W + xhi) * NCH + lc);

    // Fire all four 128-bit gathers asynchronously into LDS (ASYNCcnt path).
    async_load_b128_to_lds(lds_offset(&smem[px][0][lane]), p00);
    async_load_b128_to_lds(lds_offset(&smem[px][1][lane]), p01);
    async_load_b128_to_lds(lds_offset(&smem[px][2][lane]), p10);
    async_load_b128_to_lds(lds_offset(&smem[px][3][lane]), p11);

    // Wait for this wave's async transfers; each lane only reads back the
    // LDS slots it requested, so no cross-wave barrier is needed.
    asm volatile("s_wait_asynccnt 0" ::: "memory");

    const v4f v00 = smem[px][0][lane];
    const v4f v01 = smem[px][1][lane];
    const v4f v10 = smem[px][2][lane];
    const v4f v11 = smem[px][3][lane];

    const float gx = fx, hx = 1.0f - fx;
    const float gy = fy, hy = 1.0f - fy;

    v4f o;
    o.x = (v00.x * hx + v01.x * gx) * hy + (v10.x * hx + v11.x * gx) * gy;
    o.y = (v00.y * hx + v01.y * gx) * hy + (v10.y * hx + v11.y * gx) * gy;
    o.z = (v00.z * hx + v01.z * gx) * hy + (v10.z * hx + v11.z * gx) * gy;
    o.w = (v00.w * hx + v01.w * gx) * hy + (v10.w * hx + v11.w * gx) * gy;

    // out[roi][py][px][c] -- write-once stream, bypass long-term caching.
    v4f* outp = (v4f*)(out + (((size_t)roi * POOL + py) * POOL + px) * NCH + lc);
    __builtin_nontemporal_store(o, outp);
}

extern "C" void kernel_launch(void* const* d_in, const int* in_sizes, int n_in,
                              void* d_out, int out_size, void* d_ws, size_t ws_size,
                              hipStream_t stream)
{
    const float* img  = (const float*)d_in[0];   // (1,1024,1024,128) f32
    const float* rois = (const float*)d_in[1];   // (1,512,4) f32
    float* out = (float*)d_out;                  // (1,512,7,7,128) f32

    const int n_rois = in_sizes[1] / 4;
    dim3 grid(n_rois, POOL);
    roi_bilinear_async_kernel<<<grid, POOL * 32, 0, stream>>>(img, rois, out);
}